// YOLOLoss_13700945674420
// MI455X (gfx1250) — compile-verified
//
#include <hip/hip_runtime.h>
#include <math.h>

// Problem constants (from reference)
#define B_      32
#define A_      3
#define S_      80
#define NC_     80
#define NCELLS  (B_*A_*S_*S_)     // 614400
#define NGROUPS (NCELLS/32)       // 19200 (exact)
#define PSTR    85
#define TSTR    6
#define EPSF    1e-7f
#define FOUR_OVER_PI2 0.40528473456935109f

#define NB_BLOCKS 1024
#define BLOCK     256
#define WPB       (BLOCK/32)      // 8 waves per block (wave32)

typedef __attribute__((ext_vector_type(2))) float v2f;
typedef __attribute__((ext_vector_type(8))) float v8f;

// Full-wave (32-lane) f32 sum on the matrix unit.
// A-matrix 16x4: VGPR0 = x  -> A[m][0]=x[m], A[m][2]=x[m+16]; VGPR1 = 0.
// B-matrix 4x16 all ones    -> D[m][n] = x[m] + x[m+16]  (independent of n).
// Per-lane sum of the 8 D VGPRs gives half-sums; one shfl_xor(16) completes it.
// Requires EXEC == all ones (guaranteed at all call sites: wave-uniform control flow).
__device__ __forceinline__ float wave_sum_wmma(float x) {
  v2f a; a[0] = x;   a[1] = 0.0f;
  v2f b; b[0] = 1.0f; b[1] = 1.0f;
  v8f c = {};
  v8f d = __builtin_amdgcn_wmma_f32_16x16x4_f32(false, a, false, b, (short)0, c,
                                                false, false);
  float s = ((d[0] + d[1]) + (d[2] + d[3])) + ((d[4] + d[5]) + (d[6] + d[7]));
  s += __shfl_xor(s, 16, 32);
  return s;   // uniform across all 32 lanes
}

// Read one lane's float via v_readlane (result is scalar/SGPR). srcLane must be
// wave-uniform; EXEC is full at all call sites.
__device__ __forceinline__ float read_lane_f32(float v, int srcLane) {
  return __int_as_float(__builtin_amdgcn_readlane(__float_as_int(v), srcLane));
}

__device__ __forceinline__ float bce_logits(float z, float t) {
  // max(z,0) - z*t + log1p(exp(-|z|))
  return fmaxf(z, 0.0f) - z * t + log1pf(expf(-fabsf(z)));
}

__global__ __launch_bounds__(BLOCK)
void yolo_partial(const float* __restrict__ pred,
                  const float* __restrict__ target,
                  const float* __restrict__ anchors,
                  float* __restrict__ part) {
  const int lane   = threadIdx.x & 31;
  const int wib    = threadIdx.x >> 5;
  const int gwave  = (blockIdx.x * BLOCK + threadIdx.x) >> 5;
  const int nwaves = NB_BLOCKS * WPB;

  // Anchors: 6 uniform scalars -> scalar loads
  const float anc0 = anchors[0], anc1 = anchors[1], anc2 = anchors[2],
              anc3 = anchors[3], anc4 = anchors[4], anc5 = anchors[5];

  float accNoObj = 0.f, accNNo = 0.f, accNObj = 0.f;
  float accBox = 0.f, accObjL = 0.f, accCls = 0.f;

  for (int g = gwave; g < NGROUPS; g += nwaves) {        // wave-uniform loop
    const int  cell = g * 32 + lane;                     // one cell per lane
    const long pb   = (long)cell * PSTR;
    const long tb   = (long)cell * TSTR;

    const float tobj = __builtin_nontemporal_load(&target[tb]);
    const float z0   = __builtin_nontemporal_load(&pred[pb]);
    const bool isObj = (tobj == 1.0f);
    const bool isNo  = (tobj == 0.0f);

    // --- cheap path: every cell ---
    accNoObj += isNo  ? bce_logits(z0, tobj) : 0.f;
    accNNo   += isNo  ? 1.f : 0.f;
    accNObj  += isObj ? 1.f : 0.f;

    // --- object-only lane-parallel path: box CIoU + object BCE ---
    float tcls = 0.f;
    if (isObj) {
      const float z1 = __builtin_nontemporal_load(&pred[pb + 1]);
      const float z2 = __builtin_nontemporal_load(&pred[pb + 2]);
      const float z3 = __builtin_nontemporal_load(&pred[pb + 3]);
      const float z4 = __builtin_nontemporal_load(&pred[pb + 4]);
      const float t1 = target[tb + 1], t2 = target[tb + 2];
      const float t3 = target[tb + 3], t4 = target[tb + 4];
      tcls = target[tb + 5];

      const int   a  = (cell / (S_ * S_)) % A_;
      const float aw = (a == 0) ? anc0 : ((a == 1) ? anc2 : anc4);
      const float ah = (a == 0) ? anc1 : ((a == 1) ? anc3 : anc5);

      const float px = 1.f / (1.f + expf(-z1));
      const float py = 1.f / (1.f + expf(-z2));
      const float pw = expf(z3) * aw;
      const float ph = expf(z4) * ah;

      const float x1a = px - pw * 0.5f, x1b = px + pw * 0.5f;
      const float y1a = py - ph * 0.5f, y1b = py + ph * 0.5f;
      const float x2a = t1 - t3 * 0.5f, x2b = t1 + t3 * 0.5f;
      const float y2a = t2 - t4 * 0.5f, y2b = t2 + t4 * 0.5f;
      const float iw  = fmaxf(fminf(x1b, x2b) - fmaxf(x1a, x2a), 0.f);
      const float ih  = fmaxf(fminf(y1b, y2b) - fmaxf(y1a, y2a), 0.f);
      const float inter = iw * ih;
      const float uni   = pw * ph + t3 * t4 - inter + EPSF;
      const float iou   = inter / uni;
      const float cw    = fmaxf(x1b, x2b) - fminf(x1a, x2a);
      const float chh   = fmaxf(y1b, y2b) - fminf(y1a, y2a);
      const float c2    = cw * cw + chh * chh + EPSF;
      const float rho2  = (t1 - px) * (t1 - px) + (t2 - py) * (t2 - py);
      const float dv    = atanf(t3 / (t4 + EPSF)) - atanf(pw / (ph + EPSF));
      const float v     = FOUR_OVER_PI2 * dv * dv;
      const float alpha = v / (1.f - iou + v + EPSF);
      const float ciou  = iou - rho2 / c2 - alpha * v;
      accBox += 1.f - ciou;

      const float q = 1.f / (1.f + expf(-z0));           // reference quirk:
      accObjL += bce_logits(q, tobj);                    // sigmoid into BCE-with-logits
    }

    // --- object cells: wave-cooperative 80-class log-softmax ---
    unsigned long long m = __ballot(isObj);
    while (m) {                                          // wave-uniform loop
      const int  c   = (int)__builtin_ctzll(m);          // uniform (from ballot)
      m &= (m - 1);
      const long pb2 = (long)(g * 32 + c) * PSTR;

      // lanes own channels [lane], [32+lane], [64+lane] (coalesced)
      const float p0v = __builtin_nontemporal_load(&pred[pb2 + lane]);
      const float p1v = __builtin_nontemporal_load(&pred[pb2 + 32 + lane]);
      float p2v = 0.f;
      if (lane < 21) p2v = __builtin_nontemporal_load(&pred[pb2 + 64 + lane]);

      // classes are channels 5..84
      float mx = (lane >= 5) ? p0v : -3.0e38f;
      mx = fmaxf(mx, p1v);
      if (lane < 21) mx = fmaxf(mx, p2v);
      for (int i = 16; i; i >>= 1) mx = fmaxf(mx, __shfl_xor(mx, i, 32));

      float e = expf(p1v - mx);
      if (lane >= 5) e += expf(p0v - mx);
      if (lane < 21) e += expf(p2v - mx);
      const float sumexp = wave_sum_wmma(e);             // matrix-pipe reduction

      float sl = p1v;
      if (lane >= 5) sl += p0v;
      if (lane < 21) sl += p2v;
      const float sumLogits = wave_sum_wmma(sl);

      const float logZ = mx + logf(sumexp);

      // target class of cell c: v_readlane -> scalar, so the channel select
      // below becomes scalar branches (no EXEC juggling, no bpermute chain)
      const float tc  = read_lane_f32(tcls, c);
      const int   chY = 5 + (int)tc;                     // scalar, in [5,84]
      float ly;
      if (chY < 32)      ly = read_lane_f32(p0v, chY);
      else if (chY < 64) ly = read_lane_f32(p1v, chY - 32);
      else               ly = read_lane_f32(p2v, chY - 64);

      const float ce = -(0.9f * (ly - logZ) +
                         (0.1f / (float)NC_) * (sumLogits - (float)NC_ * logZ));
      accCls += (lane == 0) ? ce : 0.f;
    }
  }

  // Per-wave reductions of all six accumulators via WMMA
  const float rNoObj = wave_sum_wmma(accNoObj);
  const float rNNo   = wave_sum_wmma(accNNo);
  const float rNObj  = wave_sum_wmma(accNObj);
  const float rBox   = wave_sum_wmma(accBox);
  const float rObjL  = wave_sum_wmma(accObjL);
  const float rCls   = wave_sum_wmma(accCls);

  __shared__ float sh[WPB][6];
  if (lane == 0) {
    sh[wib][0] = rNoObj; sh[wib][1] = rNNo;  sh[wib][2] = rNObj;
    sh[wib][3] = rBox;   sh[wib][4] = rObjL; sh[wib][5] = rCls;
  }
  __syncthreads();
  if (threadIdx.x < 6) {
    float s = 0.f;
    for (int w = 0; w < WPB; ++w) s += sh[w][threadIdx.x];   // fixed order
    part[blockIdx.x * 6 + threadIdx.x] = s;                  // overwritten every launch
  }
}

__global__ __launch_bounds__(256)
void yolo_final(const float* __restrict__ part, float* __restrict__ out) {
  __shared__ float sh[256][6];
  float l[6] = {0.f, 0.f, 0.f, 0.f, 0.f, 0.f};
  for (int b = threadIdx.x; b < NB_BLOCKS; b += 256)
    for (int j = 0; j < 6; ++j) l[j] += part[b * 6 + j];
  for (int j = 0; j < 6; ++j) sh[threadIdx.x][j] = l[j];
  __syncthreads();
  for (int s = 128; s > 0; s >>= 1) {                        // fixed tree: deterministic
    if (threadIdx.x < s)
      for (int j = 0; j < 6; ++j) sh[threadIdx.x][j] += sh[threadIdx.x + s][j];
    __syncthreads();
  }
  if (threadIdx.x == 0) {
    const float SnoObj = sh[0][0], Nno  = sh[0][1], Nobj = sh[0][2];
    const float Sbox   = sh[0][3], SobjL = sh[0][4], Scls = sh[0][5];
    const float no_object_loss = SnoObj / fmaxf(Nno, 1.f);
    const float box_loss       = Sbox  / fmaxf(Nobj, 1.f);
    const float object_loss    = SobjL / fmaxf(Nobj, 1.f);
    const float class_loss     = Scls  / fmaxf(Nobj, 1.f);
    out[0] = 2.f * box_loss + object_loss + no_object_loss + class_loss;
  }
}

extern "C" void kernel_launch(void* const* d_in, const int* in_sizes, int n_in,
                              void* d_out, int out_size, void* d_ws, size_t ws_size,
                              hipStream_t stream) {
  (void)in_sizes; (void)n_in; (void)out_size; (void)ws_size;
  const float* pred    = (const float*)d_in[0];
  const float* target  = (const float*)d_in[1];
  const float* anchors = (const float*)d_in[2];
  float* part = (float*)d_ws;            // NB_BLOCKS*6 floats = 24 KB scratch

  yolo_partial<<<NB_BLOCKS, BLOCK, 0, stream>>>(pred, target, anchors, part);
  yolo_final<<<1, 256, 0, stream>>>(part, (float*)d_out);
}